// AttentionFlowLayer_8005819040261
// MI455X (gfx1250) — compile-verified
//
#include <hip/hip_runtime.h>

typedef __attribute__((ext_vector_type(2))) float v2f;
typedef __attribute__((ext_vector_type(8))) float v8f;

#define D_MODEL 128
#define T_LEN   2048
#define J_LEN   512
#define NTILES  4   // 4 N-tiles of 16 -> 64 columns per wave

// One wave32 computes a 16x64 output strip of S[b] = p1 + p2 + (C*wM)@Q^T + bias
// using V_WMMA_F32_16X16X4_F32 over K=128 (32 k-steps), with the rank-1 terms
// (p1 = C.wC per row, p2 = Q.wQ per col) fused into the fragment-load path.
__global__ __launch_bounds__(32) void aflow_wmma_f32_kernel(
    const float* __restrict__ C,    // (B, 2048, 128)
    const float* __restrict__ Q,    // (B, 512, 128)
    const float* __restrict__ wC,   // (128,)
    const float* __restrict__ wQ,   // (128,)
    const float* __restrict__ wM,   // (128,)  weight_CmQ
    const float* __restrict__ bias, // (1,)
    float* __restrict__ out)        // (B, 2048, 512)
{
    const int lane = threadIdx.x;        // 0..31
    const int half = lane >> 4;          // 0: K={0,1} mod 4, 1: K={2,3} mod 4
    const int lm   = lane & 15;          // row (A) / col (B) within tile
    const int b    = blockIdx.z;
    const int t0   = blockIdx.y * 16;
    const int j0   = blockIdx.x * (16 * NTILES);

    const float* Crow  = C + ((size_t)b * T_LEN + (t0 + lm)) * D_MODEL;
    const float* Qbase = Q + ((size_t)b * J_LEN + (j0 + lm)) * D_MODEL;
    const int koff = half * 2;

    v8f acc[NTILES];
#pragma unroll
    for (int n = 0; n < NTILES; ++n) acc[n] = (v8f){0,0,0,0,0,0,0,0};

    float p1p = 0.0f;                 // partial of dot(C[t0+lm,:], wC) over this lane's K-half
    float p2p[NTILES] = {0,0,0,0};    // partial of dot(Q[j,:], wQ) for this lane's column

#pragma unroll 4
    for (int k0 = 0; k0 < D_MODEL; k0 += 4) {
        const int k = k0 + koff;
        // A fragment source: raw C values (also feed p1), then scale by wCmQ for WMMA.
        v2f c2  = *(const v2f*)(Crow + k);
        v2f wc2 = *(const v2f*)(wC + k);
        v2f wm2 = *(const v2f*)(wM + k);
        v2f wq2 = *(const v2f*)(wQ + k);

        p1p += c2.x * wc2.x + c2.y * wc2.y;

        v2f a;
        a.x = c2.x * wm2.x;
        a.y = c2.y * wm2.y;

#pragma unroll
        for (int n = 0; n < NTILES; ++n) {
            // B fragment: B[k][col] = Q[col][k]; lane lm holds column lm of the tile.
            v2f q2 = *(const v2f*)(Qbase + (size_t)n * 16 * D_MODEL + k);
            p2p[n] += q2.x * wq2.x + q2.y * wq2.y;
            // D = A x B + C   (f32 16x16x4)
            acc[n] = __builtin_amdgcn_wmma_f32_16x16x4_f32(
                /*neg_a=*/false, a, /*neg_b=*/false, q2,
                /*c_mod=*/(short)0, acc[n], /*reuse_a=*/false, /*reuse_b=*/false);
        }
    }

    // Combine the two K-halves of the rank-1 dot products.
    float p1f = p1p + __shfl_xor(p1p, 16);   // lanes l and l^16 both hold p1[t0+lm]
    const float b0 = bias[0];

#pragma unroll
    for (int n = 0; n < NTILES; ++n) {
        float p2f = p2p[n] + __shfl_xor(p2p[n], 16);   // p2 for column j0+n*16+lm
        float* obase = out + ((size_t)b * T_LEN + t0) * J_LEN + j0 + n * 16 + lm;
#pragma unroll
        for (int i = 0; i < 8; ++i) {
            const int m = i + 8 * half;                // C/D layout: VGPR i -> row i (+8 upper half)
            float p1v = __shfl(p1f, m);                // broadcast p1[t0+m]
            obase[(size_t)m * J_LEN] = acc[n][i] + p1v + p2f + b0;
        }
    }
}

extern "C" void kernel_launch(void* const* d_in, const int* in_sizes, int n_in,
                              void* d_out, int out_size, void* d_ws, size_t ws_size,
                              hipStream_t stream) {
    const float* C    = (const float*)d_in[0];
    const float* Q    = (const float*)d_in[1];
    const float* wC   = (const float*)d_in[2];
    const float* wQ   = (const float*)d_in[3];
    const float* wM   = (const float*)d_in[4];
    const float* bias = (const float*)d_in[5];
    float* out        = (float*)d_out;

    const int B = in_sizes[0] / (T_LEN * D_MODEL);   // 64

    dim3 grid(J_LEN / (16 * NTILES), T_LEN / 16, B); // (8, 128, 64)
    dim3 block(32, 1, 1);                            // one wave32 per block
    aflow_wmma_f32_kernel<<<grid, block, 0, stream>>>(C, Q, wC, wQ, wM, bias, out);
}